// SpectralConv2d_3977139716491
// MI455X (gfx1250) — compile-verified
//
#include <hip/hip_runtime.h>

// Problem sizes (fixed by the reference)
#define B_   16
#define CI_  32
#define CO_  32
#define H_   256
#define W_   256
#define M1_  32
#define M2_  32
#define NMODE (M1_ * M2_)          // 1024
#define IMG   (H_ * W_)            // 65536

// LDS row strides, padded for conflict-free WMMA fragment access (64 banks)
#define XS_LD 132                  // staged x block:  128 cols + 4 pad (== 4 mod 64)
#define TT_LD 260                  // Tt / staged Acas: 256 cols + 4 pad (== 4 mod 64)
#define US_LD 36                   // Us / staged AcasT / staged Zt: 32 cols + 4 pad

typedef __attribute__((ext_vector_type(2))) float v2f;
typedef __attribute__((ext_vector_type(4))) float v4f;
typedef __attribute__((ext_vector_type(8))) float v8f;

// ---------------------------------------------------------------------------
// V_WMMA_F32_16X16X4_F32 fragment helpers (wave32).
//
// A (16x4, MxK): lane l holds float2 {A[l&15][2*(l>>4)], A[l&15][2*(l>>4)+1]}
// B (4x16, KxN): lane l holds float2 {B[2*(l>>4)][l&15], B[2*(l>>4)+1][l&15]}
//   -> for a matrix stored row-major as [N][K] this is the same access pattern
//      as the A fragment, so one loader serves both (A from [M][K] storage,
//      B from [N][K] storage).
// C/D (16x16 f32): VGPR r, lane l -> element (row = r + 8*(l>>4), col = l&15)
// ---------------------------------------------------------------------------
__device__ __forceinline__ v2f frag_ld(const float* p, int ld, int lane, int kk) {
  return *(const v2f*)(p + (lane & 15) * ld + kk + ((lane >> 4) << 1));
}

__device__ __forceinline__ v8f wmma4(v2f a, v2f b, v8f c) {
  // (neg_a, A, neg_b, B, c_mod, C, reuse_a, reuse_b)
  return __builtin_amdgcn_wmma_f32_16x16x4_f32(false, a, false, b, (short)0, c,
                                               false, false);
}

// ---------------------------------------------------------------------------
// Kernel 1: cas tables.  Acas[k][n] = cos(2*pi*k*n/256) + sin(2*pi*k*n/256),
// k in [0,32), n in [0,256).  AcasT is the [n][k] transpose.
// ---------------------------------------------------------------------------
__global__ void prep_cas(float* __restrict__ Acas, float* __restrict__ AcasT) {
  int idx = blockIdx.x * blockDim.x + threadIdx.x;
  if (idx >= M1_ * W_) return;
  int k = idx >> 8;
  int n = idx & 255;
  int m = (k * n) & 255;                           // exact periodic reduction
  float ang = 0.02454369260617025967f * (float)m;  // 2*pi/256
  float v = cosf(ang) + sinf(ang);
  Acas[k * W_ + n] = v;
  AcasT[n * M1_ + k] = v;
}

// ---------------------------------------------------------------------------
// Kernel 2: symmetrized weights, re-laid-out mode-major and [o][i]-transposed
// so they can feed WMMA B fragments directly.
//   WeT[mode][o][i] = 0.5*(w[i][o][x][y] + w[i][o][(-x)%32][(-y)%32])
//   WoT[mode][o][i] = 0.5*(w[i][o][x][y] - w[i][o][(-x)%32][(-y)%32])
// ---------------------------------------------------------------------------
__global__ void prep_w(const float* __restrict__ w, float* __restrict__ WeT,
                       float* __restrict__ WoT) {
  int idx = blockIdx.x * blockDim.x + threadIdx.x;
  if (idx >= CI_ * CO_ * NMODE) return;
  int y = idx & 31;
  int x = (idx >> 5) & 31;
  int o = (idx >> 10) & 31;
  int i = idx >> 15;
  float a = w[idx];
  int fx = (M1_ - x) & 31;
  int fy = (M2_ - y) & 31;
  float b = w[((i * CO_ + o) * M1_ + fx) * M2_ + fy];
  int mode = x * M2_ + y;
  size_t dst = (size_t)mode * (CI_ * CO_) + o * CI_ + i;
  WeT[dst] = 0.5f * (a + b);
  WoT[dst] = 0.5f * (a - b);
}

// ---------------------------------------------------------------------------
// Kernel 3: truncated forward DHT per (b,ci):  Xm = Acas * x * Acas^T.
// The cas table (hot B operand) is staged once per WG into padded LDS, so the
// steady-state inner loop is WMMA + conflict-free LDS fragment loads only.
// Pass 1: T[h][k2] = sum_w x[h][w]*Acas[k2][w]  (store T^T in padded LDS)
//   Each wave stages its 16x128 x sub-block into LDS with fully coalesced
//   b128 loads (one x row per instruction).  x is read from HBM exactly once.
// Pass 2: Xm[k1][k2] = sum_h Acas[k1][h]*T[h][k2] -> XmT[mode][b][ci]
// ---------------------------------------------------------------------------
__global__ void fwd_dht(const float* __restrict__ x, const float* __restrict__ Acas,
                        float* __restrict__ XmT) {
  __shared__ float Ac[M1_ * TT_LD];              // staged Acas [k][n], 33.3 KB
  __shared__ float Tt[M1_ * TT_LD];              // [k2][h] padded, 33.3 KB
  __shared__ float Xs[8 * 16 * XS_LD];           // per-wave 16x128 stage, 67.6 KB
  int bci = blockIdx.x;                          // b*CI + ci
  const float* xb = x + (size_t)bci * IMG;
  int tid = threadIdx.x;
  int lane = tid & 31;
  int wave = tid >> 5;
  float* xs = Xs + wave * (16 * XS_LD);
  int rowbase = wave * 32;                       // 32 rows of x per wave

  // stage Acas (8192 floats) into padded Ac: coalesced flat b128 loads
#pragma unroll
  for (int k = 0; k < 8; ++k) {
    int f = (tid + k * 256) * 4;                 // flat float index, 16B aligned
    v4f t = *(const v4f*)(Acas + f);
    int row = f >> 8;
    int col = f & 255;
    *(v4f*)(Ac + row * TT_LD + col) = t;
  }
  __syncthreads();

  // ---- pass 1 ----
  for (int mt = 0; mt < 2; ++mt) {
    const float* xrow = xb + (size_t)(rowbase + mt * 16) * W_;
    v8f acc0 = {};
    v8f acc1 = {};
    for (int kh = 0; kh < 2; ++kh) {             // K split: cols [128*kh, 128*kh+128)
      // stage 16 rows x 128 cols (coalesced; LDS ops are in-order per wave)
#pragma unroll
      for (int r = 0; r < 16; ++r) {
        v4f t = *(const v4f*)(xrow + r * W_ + kh * 128 + lane * 4);
        *(v4f*)(xs + r * XS_LD + lane * 4) = t;
      }
#pragma unroll 8
      for (int kk = 0; kk < 128; kk += 4) {
        v2f a  = frag_ld(xs, XS_LD, lane, kk);                         // x rows
        v2f b0 = frag_ld(Ac + kh * 128, TT_LD, lane, kk);              // k2 = 0..15
        v2f b1 = frag_ld(Ac + 16 * TT_LD + kh * 128, TT_LD, lane, kk); // k2 = 16..31
        acc0 = wmma4(a, b0, acc0);
        acc1 = wmma4(a, b1, acc1);
      }
    }
    int col = lane & 15;
    int rb = (lane >> 4) << 3;
#pragma unroll
    for (int r = 0; r < 8; ++r) {
      int h = rowbase + mt * 16 + rb + r;
      Tt[col * TT_LD + h]        = acc0[r];
      Tt[(16 + col) * TT_LD + h] = acc1[r];
    }
  }
  __syncthreads();

  // ---- pass 2 (waves 0..3, one 16x16 tile each) ----
  if (wave < 4) {
    int mt = wave >> 1;
    int nt = wave & 1;
    v8f acc = {};
#pragma unroll 8
    for (int kk = 0; kk < H_; kk += 4) {
      v2f a = frag_ld(Ac + mt * 16 * TT_LD, TT_LD, lane, kk);  // Acas rows k1
      v2f b = frag_ld(Tt + nt * 16 * TT_LD, TT_LD, lane, kk);  // T^T rows k2
      acc = wmma4(a, b, acc);
    }
    int col = lane & 15;
    int rb = (lane >> 4) << 3;
#pragma unroll
    for (int r = 0; r < 8; ++r) {
      int k1 = mt * 16 + rb + r;
      int k2 = nt * 16 + col;
      XmT[(size_t)(k1 * M2_ + k2) * (B_ * CI_) + bci] = acc[r];
    }
  }
}

// ---------------------------------------------------------------------------
// Kernel 4: per-mode channel mixing.  One wave per mode:
//   Z(16x32) = X(16x32) @ We(32x32) + Xflip(16x32) @ Wo(32x32)
// All operands are contiguous, L2-resident blocks (12 MB workspace << 192 MB
// L2).  Writes Z transposed per (b,co): Zt[b][co][k2][k1] for the inverse.
// ---------------------------------------------------------------------------
__global__ void mode_mix(const float* __restrict__ XmT, const float* __restrict__ WeT,
                         const float* __restrict__ WoT, float* __restrict__ Zt) {
  int gwave = (blockIdx.x * blockDim.x + threadIdx.x) >> 5;   // mode id, 0..1023
  int lane = threadIdx.x & 31;
  int k1 = gwave >> 5;
  int k2 = gwave & 31;
  int fmode = ((M1_ - k1) & 31) * M2_ + ((M2_ - k2) & 31);
  const float* X  = XmT + (size_t)gwave * (B_ * CI_);    // [b][i], ld 32
  const float* Xf = XmT + (size_t)fmode * (B_ * CI_);
  const float* We = WeT + (size_t)gwave * (CI_ * CO_);   // [o][i], ld 32
  const float* Wo = WoT + (size_t)gwave * (CI_ * CO_);

  for (int nt = 0; nt < 2; ++nt) {
    v8f acc = {};
#pragma unroll
    for (int kk = 0; kk < CI_; kk += 4) {
      v2f a  = frag_ld(X, CI_, lane, kk);
      v2f bw = frag_ld(We + nt * 16 * CI_, CI_, lane, kk);
      acc = wmma4(a, bw, acc);
      v2f af = frag_ld(Xf, CI_, lane, kk);
      v2f bo = frag_ld(Wo + nt * 16 * CI_, CI_, lane, kk);
      acc = wmma4(af, bo, acc);
    }
    int col = lane & 15;
    int rb = (lane >> 4) << 3;
#pragma unroll
    for (int r = 0; r < 8; ++r) {
      int b = rb + r;                       // 0..15 (batch)
      int co = nt * 16 + col;
      Zt[(size_t)(b * CO_ + co) * NMODE + k2 * M1_ + k1] = acc[r];
    }
  }
}

// ---------------------------------------------------------------------------
// Kernel 5: inverse DHT per (b,co):  out = (1/65536) * Acas^T * Z * Acas.
// Both hot fragment operands (AcasT and this WG's 4KB Z block) are staged into
// padded LDS once, so inner loops are WMMA + conflict-free LDS only.
// Pass 1: U[h][k2] = sum_k1 Acas[k1][h]*Z[k1][k2]   (U in padded LDS, [h][k2])
// Pass 2: out[h][w] = (1/65536) * sum_k2 U[h][k2]*Acas[k2][w]
// ---------------------------------------------------------------------------
__global__ void inv_dht(const float* __restrict__ Zt, const float* __restrict__ AcasT,
                        float* __restrict__ out) {
  __shared__ float Us[H_ * US_LD];               // [h][k2] padded, 36.9 KB
  __shared__ float As[H_ * US_LD];               // staged AcasT [n][k], 36.9 KB
  __shared__ float Zs[M2_ * US_LD];              // staged Z block [k2][k1], 4.6 KB
  int bco = blockIdx.x;
  float* ob = out + (size_t)bco * IMG;
  int tid = threadIdx.x;
  int lane = tid & 31;
  int wave = tid >> 5;
  int rowbase = wave * 32;

  // stage AcasT (8192 floats) into padded As: coalesced flat b128 loads
#pragma unroll
  for (int k = 0; k < 8; ++k) {
    int f = (tid + k * 256) * 4;                 // flat float index, 16B aligned
    v4f t = *(const v4f*)(AcasT + f);
    int row = f >> 5;
    int col = f & 31;
    *(v4f*)(As + row * US_LD + col) = t;
  }
  // stage this (b,co)'s Z block (1024 floats) into padded Zs
  {
    int f = tid * 4;
    v4f t = *(const v4f*)(Zt + (size_t)bco * NMODE + f);
    *(v4f*)(Zs + (f >> 5) * US_LD + (f & 31)) = t;
  }
  __syncthreads();

  // ---- pass 1 ----
  for (int mt = 0; mt < 2; ++mt) {
    const float* arow = As + (rowbase + mt * 16) * US_LD;   // [h][k1]
    for (int nt = 0; nt < 2; ++nt) {
      v8f acc = {};
#pragma unroll
      for (int kk = 0; kk < M1_; kk += 4) {
        v2f a = frag_ld(arow, US_LD, lane, kk);
        v2f b = frag_ld(Zs + nt * 16 * US_LD, US_LD, lane, kk);
        acc = wmma4(a, b, acc);
      }
      int col = lane & 15;
      int rb = (lane >> 4) << 3;
#pragma unroll
      for (int r = 0; r < 8; ++r) {
        int h = rowbase + mt * 16 + rb + r;
        Us[h * US_LD + nt * 16 + col] = acc[r];
      }
    }
  }
  __syncthreads();

  // ---- pass 2 ----
  const float scale = 1.0f / (float)IMG;
  for (int mt = 0; mt < 2; ++mt) {
    const float* arow = Us + (rowbase + mt * 16) * US_LD;   // [h][k2]
    for (int nt = 0; nt < 16; ++nt) {
      v8f acc = {};
#pragma unroll
      for (int kk = 0; kk < M2_; kk += 4) {
        v2f a = frag_ld(arow, US_LD, lane, kk);
        v2f b = frag_ld(As + nt * 16 * US_LD, US_LD, lane, kk);  // [w][k2]
        acc = wmma4(a, b, acc);
      }
      int col = lane & 15;
      int rb = (lane >> 4) << 3;
#pragma unroll
      for (int r = 0; r < 8; ++r) {
        int h = rowbase + mt * 16 + rb + r;
        ob[h * W_ + nt * 16 + col] = acc[r] * scale;
      }
    }
  }
}

// ---------------------------------------------------------------------------
extern "C" void kernel_launch(void* const* d_in, const int* in_sizes, int n_in,
                              void* d_out, int out_size, void* d_ws, size_t ws_size,
                              hipStream_t stream) {
  const float* x  = (const float*)d_in[0];   // (16,32,256,256) f32
  const float* w1 = (const float*)d_in[1];   // (32,32,32,32)   f32
  float* out = (float*)d_out;                // (16,32,256,256) f32
  float* ws = (float*)d_ws;

  // workspace layout (floats): ~12.1 MB total (fully L2-resident intermediates)
  float* Acas  = ws;                          // 32*256
  float* AcasT = Acas + M1_ * W_;             // 256*32
  float* WeT   = AcasT + H_ * M1_;            // 1024*32*32
  float* WoT   = WeT + (size_t)NMODE * CI_ * CO_;
  float* XmT   = WoT + (size_t)NMODE * CI_ * CO_;   // 1024*16*32
  float* Zt    = XmT + (size_t)NMODE * B_ * CI_;    // 16*32*1024

  prep_cas<<<(M1_ * W_ + 255) / 256, 256, 0, stream>>>(Acas, AcasT);
  prep_w<<<(CI_ * CO_ * NMODE + 255) / 256, 256, 0, stream>>>(w1, WeT, WoT);
  fwd_dht<<<B_ * CI_, 256, 0, stream>>>(x, Acas, XmT);
  mode_mix<<<(NMODE * 32) / 256, 256, 0, stream>>>(XmT, WeT, WoT, Zt);
  inv_dht<<<B_ * CO_, 256, 0, stream>>>(Zt, AcasT, out);
}